// YoloV3Loss_36344013259292
// MI455X (gfx1250) — compile-verified
//
#include <hip/hip_runtime.h>
#include <math.h>

// f32 ext-vector types for the f32 WMMA path (16x16x4: A/B = 2 VGPRs, C/D = 8 VGPRs)
typedef float vf2 __attribute__((ext_vector_type(2)));
typedef float vf8 __attribute__((ext_vector_type(8)));

#define THREADS 256
#define WAVES_PER_BLOCK (THREADS / 32)

__device__ __forceinline__ float bce_logits(float x, float t) {
    // max(x,0) - x*t + log1p(exp(-|x|)) : numerically stable BCEWithLogits
    return fmaxf(x, 0.0f) - x * t + log1pf(expf(-fabsf(x)));
}

// Pass 1: per-cell loss terms, block-level reduction to float4 partials
// (sum_noobj, count_noobj, sum_obj, count_obj)
__global__ __launch_bounds__(THREADS)
void yolo_loss_partial(const float* __restrict__ pred,
                       const float* __restrict__ tgt,
                       const float* __restrict__ anchors,
                       float4* __restrict__ partials,
                       int N, int S, int A)
{
    float s_no = 0.0f, c_no = 0.0f, s_ob = 0.0f, c_ob = 0.0f;
    const int stride = gridDim.x * blockDim.x;

    for (int cell = blockIdx.x * blockDim.x + threadIdx.x; cell < N; cell += stride) {
        // targets: 6-float records, 24B -> always 8B aligned: two b64 loads + one b32 load (use 2.5)
        const vf2* trec = (const vf2*)tgt + 3 * (size_t)cell;
        vf2 t01 = trec[0];                       // t0 = obj flag, t1 = cx offset
        vf2 t23 = trec[1];                       // t2 = cy offset, t3 = w
        float t4 = tgt[6 * (size_t)cell + 4];    // t4 = h
        // predictions channel 0 gather (stride 340B); one line per cell, L2-resident on replay
        float p = pred[85 * (size_t)cell];

        float t0 = t01.x;
        bool  noobj = (t0 == 0.0f);
        bool  obj   = (t0 == 1.0f);

        if (noobj) {
            s_no += bce_logits(p, t0);
            c_no += 1.0f;
        }
        if (obj) {  // ~2% of cells take the transcendental-heavy path
            int j = cell % S;               // grid dim-3 index
            int i = (cell / S) % S;         // grid dim-2 index
            int a = (cell / (S * S)) % A;   // anchor index
            float ax = anchors[2 * a + 0];
            float ay = anchors[2 * a + 1];

            float cx = t01.y, cy = t23.x, w = t23.y, h = t4;
            float bx = 1.0f / (1.0f + expf(-cx)) + (float)i;   // sigmoid(t1) + cells_i
            float by = 1.0f / (1.0f + expf(-cy)) + (float)j;   // sigmoid(t2) + cells_j
            float bw = expf(w) * ax;
            float bh = expf(h) * ay;

            // elementwise IoU in (cx,cy,w,h) format
            float x1a = bx - 0.5f * bw, x2a = bx + 0.5f * bw;
            float y1a = by - 0.5f * bh, y2a = by + 0.5f * bh;
            float x1b = cx - 0.5f * w,  x2b = cx + 0.5f * w;
            float y1b = cy - 0.5f * h,  y2b = cy + 0.5f * h;
            float iw = fmaxf(fminf(x2a, x2b) - fmaxf(x1a, x1b), 0.0f);
            float ih = fmaxf(fminf(y2a, y2b) - fmaxf(y1a, y1b), 0.0f);
            float inter = iw * ih;
            float uni = (x2a - x1a) * (y2a - y1a) + (x2b - x1b) * (y2b - y1b) - inter;
            float iou = inter / (uni + 1e-6f);

            // faithful to source: input = iou, target = prediction logit
            s_ob += bce_logits(iou, p);
            c_ob += 1.0f;
        }
    }

    // wave32 reduction: 5 xor-shuffle steps
    #pragma unroll
    for (int off = 16; off > 0; off >>= 1) {
        s_no += __shfl_xor(s_no, off, 32);
        c_no += __shfl_xor(c_no, off, 32);
        s_ob += __shfl_xor(s_ob, off, 32);
        c_ob += __shfl_xor(c_ob, off, 32);
    }

    __shared__ float4 lds[WAVES_PER_BLOCK];
    int wid  = threadIdx.x >> 5;
    int lane = threadIdx.x & 31;
    if (lane == 0) lds[wid] = make_float4(s_no, c_no, s_ob, c_ob);
    __syncthreads();
    if (threadIdx.x == 0) {
        float4 r = lds[0];
        #pragma unroll
        for (int w = 1; w < WAVES_PER_BLOCK; ++w) {
            r.x += lds[w].x; r.y += lds[w].y; r.z += lds[w].z; r.w += lds[w].w;
        }
        partials[blockIdx.x] = r;  // deterministic: no float atomics
    }
}

#if __has_builtin(__builtin_amdgcn_wmma_f32_16x16x4_f32)
// Wave-wide sum of one float per lane via V_WMMA_F32_16X16X4_F32.
// A = ones(16x4)  =>  D[r][j] = colsum_j(B) for every row r.
// B's 64 register slots (2 VGPRs x 32 lanes) biject onto its 64 elements, so
// putting x in slot0 and 0 in slot1 makes sum-of-all-B-elements == wave sum(x).
// d[0] holds rows {0,8} across the 32 lanes; both rows equal the colsums, so
// summing d[0] over all 32 lanes yields exactly 2 * wave_sum(x).
__device__ __forceinline__ float wave_sum_wmma(float x) {
    vf2 ones; ones.x = 1.0f; ones.y = 1.0f;
    vf2 b;    b.x = x;       b.y = 0.0f;
    vf8 c = {};
    vf8 d = __builtin_amdgcn_wmma_f32_16x16x4_f32(
        /*neg_a=*/false, ones, /*neg_b=*/false, b,
        /*c_mod=*/(short)0, c, /*reuse_a=*/false, /*reuse_b=*/false);
    float v = d[0];
    #pragma unroll
    for (int off = 16; off > 0; off >>= 1) v += __shfl_xor(v, off, 32);
    return 0.5f * v;
}
#else
__device__ __forceinline__ float wave_sum_wmma(float x) {
    #pragma unroll
    for (int off = 16; off > 0; off >>= 1) x += __shfl_xor(x, off, 32);
    return x;
}
#endif

// Pass 2: one full wave (EXEC all ones -> WMMA legal) folds block partials
__global__ __launch_bounds__(32)
void yolo_loss_final(const float4* __restrict__ partials, int G,
                     float* __restrict__ out)
{
    int lane = threadIdx.x;
    float s_no = 0.0f, c_no = 0.0f, s_ob = 0.0f, c_ob = 0.0f;
    for (int i = lane; i < G; i += 32) {
        float4 r = partials[i];
        s_no += r.x; c_no += r.y; s_ob += r.z; c_ob += r.w;
    }
    s_no = wave_sum_wmma(s_no);
    c_no = wave_sum_wmma(c_no);
    s_ob = wave_sum_wmma(s_ob);
    c_ob = wave_sum_wmma(c_ob);
    if (lane == 0) {
        float no_obj_loss = s_no / fmaxf(c_no, 1.0f);
        float obj_loss    = s_ob / fmaxf(c_ob, 1.0f);
        out[0] = 0.5f * no_obj_loss + obj_loss;
    }
}

extern "C" void kernel_launch(void* const* d_in, const int* in_sizes, int n_in,
                              void* d_out, int out_size, void* d_ws, size_t ws_size,
                              hipStream_t stream) {
    const float* pred    = (const float*)d_in[0];  // [B,A,S,S,85] f32
    const float* tgt     = (const float*)d_in[1];  // [B,A,S,S,6]  f32
    const float* anchors = (const float*)d_in[2];  // [A,2]        f32
    float*       out     = (float*)d_out;          // scalar f32

    const int S = 52, A = 3;
    int N = in_sizes[1] / 6;                       // 259,584 cells

    int blocks = (N + THREADS * 2 - 1) / (THREADS * 2);  // ~2 cells/thread -> 507 blocks
    if (blocks < 1) blocks = 1;
    size_t need = (size_t)blocks * sizeof(float4);
    if (need > ws_size && ws_size >= sizeof(float4))
        blocks = (int)(ws_size / sizeof(float4));

    float4* partials = (float4*)d_ws;
    yolo_loss_partial<<<blocks, THREADS, 0, stream>>>(pred, tgt, anchors, partials, N, S, A);
    yolo_loss_final<<<1, 32, 0, stream>>>(partials, blocks, out);
}